// VectorQuantize_55070070669391
// MI455X (gfx1250) — compile-verified
//
#include <hip/hip_runtime.h>
#include <hip/hip_bf16.h>
#include <cstdint>
#include <cstddef>

#define B_      16
#define D_IN    1024
#define T_      4096
#define CB_SZ   1024
#define CB_DIM  8
#define NTOK    (B_ * T_)   // 65536

typedef float v2f __attribute__((ext_vector_type(2)));
typedef float v8f __attribute__((ext_vector_type(8)));

// d_out layout (floats), concatenated in reference return order:
//   z_out [B, D_IN, T], commitment[B], codebook_loss[B], indices[B,T] (as float), z_e [B, CB_DIM, T]
#define OUT_COMMIT ((size_t)B_ * D_IN * T_)       // 67108864
#define OUT_CBLOSS (OUT_COMMIT + B_)
#define OUT_IDX    (OUT_CBLOSS + B_)
#define OUT_ZE     (OUT_IDX + NTOK)

__device__ __forceinline__ v8f wmma_f32_4(v2f a, v2f b, v8f c) {
#if defined(__AMDGCN__)
  return __builtin_amdgcn_wmma_f32_16x16x4_f32(false, a, false, b, (short)0, c, false, false);
#else
  return c;
#endif
}

// ---------------- weight prep ----------------
// w_in stored d-major: win_t[d*8 + o] = in_g[o] * in_v[o,d] / ||in_v[o,:]||
__global__ void __launch_bounds__(256) k_prep_win(const float* __restrict__ in_v,
                                                  const float* __restrict__ in_g,
                                                  float* __restrict__ win_t) {
  __shared__ float red[256];
  const int r = blockIdx.x;            // 0..7 output rows
  const int tid = threadIdx.x;
  float s = 0.f;
  for (int i = tid; i < D_IN; i += 256) { float v = in_v[r * D_IN + i]; s += v * v; }
  red[tid] = s; __syncthreads();
  for (int off = 128; off > 0; off >>= 1) {
    if (tid < off) red[tid] += red[tid + off];
    __syncthreads();
  }
  const float scale = in_g[r] / sqrtf(red[0]);
  for (int i = tid; i < D_IN; i += 256) win_t[i * CB_DIM + r] = in_v[r * D_IN + i] * scale;
}

// w_out[o*8+k] = out_g[o]*out_v[o,k]/||out_v[o,:]||  ;  cbn = codebook / max(||row||, eps)
__global__ void __launch_bounds__(256) k_prep_wout_cb(const float* __restrict__ out_v,
                                                      const float* __restrict__ out_g,
                                                      const float* __restrict__ cb,
                                                      float* __restrict__ wout,
                                                      float* __restrict__ cbn) {
  const int r = blockIdx.x * 256 + threadIdx.x;   // 0..2047
  if (r < CB_SZ) {
    float v[8]; float s = 0.f;
    #pragma unroll
    for (int k = 0; k < 8; k++) { v[k] = out_v[r * 8 + k]; s += v[k] * v[k]; }
    const float sc = out_g[r] / sqrtf(s);
    #pragma unroll
    for (int k = 0; k < 8; k++) wout[r * 8 + k] = v[k] * sc;
  } else {
    const int c = r - CB_SZ;
    float v[8]; float s = 0.f;
    #pragma unroll
    for (int k = 0; k < 8; k++) { v[k] = cb[c * 8 + k]; s += v[k] * v[k]; }
    const float sc = 1.f / fmaxf(sqrtf(s), 1e-12f);
    #pragma unroll
    for (int k = 0; k < 8; k++) cbn[c * 8 + k] = v[k] * sc;
  }
}

// ---------------- in_proj via WMMA ----------------
// z_e[b,o,t] = sum_d w_in[o,d] z[b,d,t] + in_b[o]
// Per wave: 16-token tile. D[M=t, N=o] = A[t,d] * B[d,o], 256 k-steps of 4.
__global__ void __launch_bounds__(256) k_in_proj(const float* __restrict__ z,
                                                 const float* __restrict__ win_t,
                                                 const float* __restrict__ in_b,
                                                 float* __restrict__ ze) {
  __shared__ float wl[D_IN * CB_DIM];   // 32 KB, d-major w_in
  const int tid = threadIdx.x;
  for (int i = tid; i < D_IN * CB_DIM; i += 256) wl[i] = win_t[i];
  __syncthreads();

  const int wave = tid >> 5, lane = tid & 31, lane16 = lane & 15;
  const bool hi = lane >= 16;
  const int l7 = lane16 & 7;            // N>=8 columns are dead; alias to valid LDS rows
  const int tile = blockIdx.x * 8 + wave;   // 4096 token tiles
  const int tok0 = tile << 4;
  const int b = tok0 >> 12;             // tile never crosses a batch (T%16==0)
  const int t0 = tok0 & 4095;

  const int kA0 = hi ? 2 : 0, kA1 = hi ? 3 : 1;
  const size_t offA0 = (size_t)kA0 * T_, offA1 = (size_t)kA1 * T_;
  const float* zp = z + (size_t)b * D_IN * T_ + t0 + lane16;

  const float ib = in_b[l7];
  v8f c;
  #pragma unroll
  for (int r = 0; r < 8; r++) c[r] = ib;

  #pragma unroll 2
  for (int d0 = 0; d0 < D_IN; d0 += 4) {
    v2f a, bb;
    a.x = __builtin_nontemporal_load(zp + (size_t)d0 * T_ + offA0);
    a.y = __builtin_nontemporal_load(zp + (size_t)d0 * T_ + offA1);
    bb.x = wl[((d0 + kA0) << 3) + l7];
    bb.y = wl[((d0 + kA1) << 3) + l7];
    c = wmma_f32_4(a, bb, c);
  }

  // D: lane16 = o (valid < 8), M = r + 8*hi = t offset
  if (lane16 < 8) {
    const int tbase = t0 + (hi ? 8 : 0);
    #pragma unroll
    for (int r = 0; r < 8; r++)
      ze[((size_t)(b * 8 + lane16)) * T_ + tbase + r] = c[r];
  }
}

// ---------------- distance GEMM + argmin via WMMA ----------------
__global__ void __launch_bounds__(256) k_argmin(const float* __restrict__ ze,
                                                const float* __restrict__ cbn,
                                                int* __restrict__ idx_i,
                                                float* __restrict__ idx_f) {
  __shared__ float cl[CB_SZ * CB_DIM];     // 32 KB normalized codebook
  const int tid = threadIdx.x;
  for (int i = tid; i < CB_SZ * CB_DIM; i += 256) cl[i] = cbn[i];
  __syncthreads();

  const int wave = tid >> 5, lane = tid & 31, lane16 = lane & 15;
  const bool hi = lane >= 16;
  const int tile = blockIdx.x * 8 + wave;   // 4096 token tiles
  const int tok16 = tile << 4;
  const int token = tok16 + lane16;         // lanes 16..31 duplicate 0..15 for A layout
  const int b = token >> 12, t = token & 4095;

  // load token vector, L2-normalize (eps-clamped)
  float e[8]; float s = 0.f;
  #pragma unroll
  for (int o = 0; o < 8; o++) { e[o] = ze[((size_t)(b * 8 + o)) * T_ + t]; s += e[o] * e[o]; }
  const float inv = 1.f / fmaxf(sqrtf(s), 1e-12f);
  #pragma unroll
  for (int o = 0; o < 8; o++) e[o] *= inv;

  v2f a0, a1;
  a0.x = hi ? e[2] : e[0];  a0.y = hi ? e[3] : e[1];
  a1.x = hi ? e[6] : e[4];  a1.y = hi ? e[7] : e[5];
  const int k0 = hi ? 2 : 0, k1 = hi ? 3 : 1;

  float best[8]; int bidx[8];
  #pragma unroll
  for (int r = 0; r < 8; r++) { best[r] = -3.4e38f; bidx[r] = 0; }

  for (int ct = 0; ct < 64; ct++) {         // 64 code tiles of 16
    const int base = ((ct << 4) + lane16) << 3;
    v2f b0, b1;
    b0.x = cl[base + k0];      b0.y = cl[base + k1];
    b1.x = cl[base + 4 + k0];  b1.y = cl[base + 4 + k1];
    v8f c = {0.f, 0.f, 0.f, 0.f, 0.f, 0.f, 0.f, 0.f};
    c = wmma_f32_4(a0, b0, c);
    c = wmma_f32_4(a1, b1, c);
    const int n = (ct << 4) + lane16;
    #pragma unroll
    for (int r = 0; r < 8; r++) {
      const float v = c[r];
      if (v > best[r]) { best[r] = v; bidx[r] = n; }   // strict > keeps earliest n
    }
  }

  // argmax reduce across the 16 lanes sharing each M (xor 1..8 stays within half)
  #pragma unroll
  for (int r = 0; r < 8; r++) {
    #pragma unroll
    for (int off = 1; off < 16; off <<= 1) {
      const float ov = __shfl_xor(best[r], off, 32);
      const int   oi = __shfl_xor(bidx[r], off, 32);
      if (ov > best[r] || (ov == best[r] && oi < bidx[r])) { best[r] = ov; bidx[r] = oi; }
    }
  }
  if (lane == 0) {
    #pragma unroll
    for (int r = 0; r < 8; r++) { const int tm = tok16 + r; idx_i[tm] = bidx[r]; idx_f[tm] = (float)bidx[r]; }
  } else if (lane == 16) {
    #pragma unroll
    for (int r = 0; r < 8; r++) { const int tm = tok16 + 8 + r; idx_i[tm] = bidx[r]; idx_f[tm] = (float)bidx[r]; }
  }
}

// ---------------- losses (commitment == codebook numerically) ----------------
__global__ void __launch_bounds__(256) k_loss_partial(const float* __restrict__ ze,
                                                      const int* __restrict__ idx_i,
                                                      const float* __restrict__ cb,
                                                      float* __restrict__ partial) {
  __shared__ float red[256];
  const int tid = threadIdx.x;
  const int tk = blockIdx.x * 256 + tid;   // 16 blocks per batch, no cross-batch blocks
  const int b = tk >> 12, t = tk & 4095;
  const int ci = idx_i[tk];
  float s = 0.f;
  #pragma unroll
  for (int o = 0; o < 8; o++) {
    const float d = ze[((size_t)(b * 8 + o)) * T_ + t] - cb[ci * 8 + o];
    s += d * d;
  }
  red[tid] = s; __syncthreads();
  for (int off = 128; off > 0; off >>= 1) {
    if (tid < off) red[tid] += red[tid + off];
    __syncthreads();
  }
  if (tid == 0) partial[blockIdx.x] = red[0];
}

__global__ void k_loss_final(const float* __restrict__ partial,
                             float* __restrict__ commit, float* __restrict__ cbloss) {
  const int b = threadIdx.x;
  if (b < B_) {
    float s = 0.f;
    for (int i = 0; i < 16; i++) s += partial[b * 16 + i];
    const float loss = s / (float)(CB_DIM * T_);
    commit[b] = loss; cbloss[b] = loss;
  }
}

// ---------------- out_proj via WMMA ----------------
// z_out[b,o,t] = sum_k w_out[o,k]*codebook[idx[b,t],k] + out_b[o]
// Per wave: one 16-token tile; B (q^T) built once, reused across 64 o-tiles.
// D[M=o, N=t] so stores are coalesced 64B segments per half-wave.
__global__ void __launch_bounds__(256) k_out_proj(const int* __restrict__ idx_i,
                                                  const float* __restrict__ cb,
                                                  const float* __restrict__ wout,
                                                  const float* __restrict__ out_b,
                                                  float* __restrict__ zout) {
  __shared__ float wl[D_IN * CB_DIM];   // 32 KB, o-major w_out
  const int tid = threadIdx.x;
  for (int i = tid; i < D_IN * CB_DIM; i += 256) wl[i] = wout[i];
  __syncthreads();

  const int wave = tid >> 5, lane = tid & 31, lane16 = lane & 15;
  const bool hi = lane >= 16;
  const int tile = blockIdx.x * 8 + wave;   // 4096 t-tiles
  const int tok0 = tile << 4;
  const int b = tok0 >> 12;
  const int t0 = tok0 & 4095;
  const int kk = hi ? 2 : 0;                // K base for this half (v0:K=kk, v1:K=kk+1)

  // B tiles (K x N = k x t): lane holds q[K][t0+lane16]
  const int ci = idx_i[tok0 + lane16];
  const float* qrow = cb + ci * 8;
  v2f b0, b1;
  b0.x = qrow[kk];     b0.y = qrow[kk + 1];
  b1.x = qrow[4 + kk]; b1.y = qrow[4 + kk + 1];

  float* outp = zout + (size_t)b * D_IN * T_ + t0 + lane16;
  const int rbase = hi ? 8 : 0;

  for (int o0 = 0; o0 < D_IN; o0 += 16) {
    // A tiles (M x K = o x k): lane holds w_out[o0+lane16][K]
    const float* wr = wl + ((o0 + lane16) << 3);
    v2f a0, a1;
    a0.x = wr[kk];     a0.y = wr[kk + 1];
    a1.x = wr[4 + kk]; a1.y = wr[4 + kk + 1];
    v8f c = {0.f, 0.f, 0.f, 0.f, 0.f, 0.f, 0.f, 0.f};
    c = wmma_f32_4(a0, b0, c);
    c = wmma_f32_4(a1, b1, c);
    // D: M = o0 + r + 8*hi, N = t0 + lane16 ; add bias at store
    #pragma unroll
    for (int r = 0; r < 8; r++) {
      const float bias = hi ? out_b[o0 + r + 8] : out_b[o0 + r];   // uniform s_loads + cndmask
      __builtin_nontemporal_store(c[r] + bias, outp + (size_t)(o0 + r + rbase) * T_);
    }
  }
}

extern "C" void kernel_launch(void* const* d_in, const int* in_sizes, int n_in,
                              void* d_out, int out_size, void* d_ws, size_t ws_size,
                              hipStream_t stream) {
  const float* z     = (const float*)d_in[0];
  const float* in_v  = (const float*)d_in[1];
  const float* in_g  = (const float*)d_in[2];
  const float* in_b  = (const float*)d_in[3];
  const float* out_v = (const float*)d_in[4];
  const float* out_g = (const float*)d_in[5];
  const float* out_b = (const float*)d_in[6];
  const float* cb    = (const float*)d_in[7];

  float* out    = (float*)d_out;
  float* zout   = out;
  float* commit = out + OUT_COMMIT;
  float* cbloss = out + OUT_CBLOSS;
  float* idx_f  = out + OUT_IDX;
  float* ze     = out + OUT_ZE;

  float* ws      = (float*)d_ws;
  float* win_t   = ws;                 // 8192 floats (d-major w_in)
  float* wout    = ws + 8192;          // 8192 (o-major w_out)
  float* cbn     = ws + 16384;         // 8192 (normalized codebook)
  float* partial = ws + 24576;         // 256
  int*   idx_i   = (int*)(ws + 24832); // 65536 ints

  k_prep_win    <<<8,   256, 0, stream>>>(in_v, in_g, win_t);
  k_prep_wout_cb<<<8,   256, 0, stream>>>(out_v, out_g, cb, wout, cbn);
  k_in_proj     <<<512, 256, 0, stream>>>(z, win_t, in_b, ze);
  k_argmin      <<<512, 256, 0, stream>>>(ze, cbn, idx_i, idx_f);
  k_loss_partial<<<256, 256, 0, stream>>>(ze, idx_i, cb, partial);
  k_loss_final  <<<1,   32,  0, stream>>>(partial, commit, cbloss);
  k_out_proj    <<<512, 256, 0, stream>>>(idx_i, cb, wout, out_b, zout);
}